// SW_MHA_11888469475599
// MI455X (gfx1250) — compile-verified
//
#include <hip/hip_runtime.h>
#include <hip/hip_bf16.h>
#include <math.h>

typedef float v2f __attribute__((ext_vector_type(2)));
typedef float v8f __attribute__((ext_vector_type(8)));

#define C_DIM 512
#define THREE_C 1536
#define NPOS 2048
#define TILE_M 128
#define TILE_N 64
#define TILE_K 64
#define LDS_STRIDE 68   // pad 64 -> 68 dwords: fragment bank = (4*row + k) % 64, conflict-free

// ---------------------------------------------------------------------------
// Kernel 1: sinusoidal positional encoding table pe[2048][512] (double trig to
// match numpy float64 computation, cast to f32 at store)
// ---------------------------------------------------------------------------
__global__ __launch_bounds__(256) void pe_kernel(float* __restrict__ pe) {
  int idx = blockIdx.x * 256 + threadIdx.x;
  if (idx >= NPOS * C_DIM) return;
  int n = idx >> 9;
  int c = idx & 511;
  double rate = pow(10000.0, -(double)(2 * (c >> 1)) / 512.0);
  double ang = (double)n * rate;
  pe[idx] = (float)((c & 1) ? cos(ang) : sin(ang));
}

// ---------------------------------------------------------------------------
// Kernel 2: fused (x + PE) @ W GEMM using V_WMMA_F32_16X16X4_F32.
// Workgroup = 256 threads = 8 waves; tile 128(M) x 64(N); K staged in 64-blocks.
// gatherMode=0: row g -> x row g (KV projection, W cols pre-offset by +512)
// gatherMode=1: row g -> x row b*2048 + 2047-32*w  (Q at window starts)
// ---------------------------------------------------------------------------
__global__ __launch_bounds__(256) void gemm_pe_wmma(
    const float* __restrict__ x, const float* __restrict__ pe,
    const float* __restrict__ w, float* __restrict__ out,
    int outStride, int gatherMode) {
  __shared__ float ldsA[TILE_M * LDS_STRIDE];           // [m][k]
  __shared__ float ldsB[TILE_N * LDS_STRIDE];           // [n][k] (transposed)

  const int tid   = threadIdx.x;
  const int lane  = tid & 31;
  const int wv    = tid >> 5;
  const int mBase = blockIdx.x * TILE_M;
  const int nBase = blockIdx.y * TILE_N;
  const int fr    = lane & 15;            // M index (A) / N index (B,C,D)
  const int khalf = (lane >> 4) << 1;     // K sub-offset: lanes 16-31 hold K+2,K+3

  v8f acc[4] = {v8f{}, v8f{}, v8f{}, v8f{}};

  for (int kb = 0; kb < C_DIM; kb += TILE_K) {
    // --- stage A block 128x64 (x + PE fused), coalesced 64-wide rows ---
#pragma unroll
    for (int i = 0; i < 32; ++i) {
      int flat = i * 256 + tid;
      int m = flat >> 6;
      int k = flat & 63;
      int g = mBase + m;
      int xr = gatherMode ? ((g >> 6) * NPOS + (2047 - 32 * (g & 63))) : g;
      int col = kb + k;
      ldsA[m * LDS_STRIDE + k] =
          x[xr * C_DIM + col] + pe[(xr & (NPOS - 1)) * C_DIM + col];
    }
    // --- stage B block 64(K)x64(N), stored transposed [n][k] ---
#pragma unroll
    for (int i = 0; i < 16; ++i) {
      int flat = i * 256 + tid;
      int k = flat >> 6;
      int n = flat & 63;
      ldsB[n * LDS_STRIDE + k] = w[(kb + k) * THREE_C + nBase + n];
    }
    __syncthreads();

    // --- 16 WMMA steps of depth 4; A fragment reused across 4 N-tiles ---
#pragma unroll
    for (int kk = 0; kk < TILE_K; kk += 4) {
      v2f a = *(const v2f*)&ldsA[(wv * 16 + fr) * LDS_STRIDE + kk + khalf];
#pragma unroll
      for (int t = 0; t < 4; ++t) {
        v2f b = *(const v2f*)&ldsB[(t * 16 + fr) * LDS_STRIDE + kk + khalf];
        acc[t] = __builtin_amdgcn_wmma_f32_16x16x4_f32(
            false, a, false, b, (short)0, acc[t], false, false);
      }
    }
    __syncthreads();
  }

  // --- epilogue: D element (lane,j) -> row 8*(lane>>4)+j, col lane&15 ---
  const int rbase = mBase + wv * 16 + ((lane >> 4) << 3);
  const int cbase = nBase + fr;
#pragma unroll
  for (int t = 0; t < 4; ++t) {
#pragma unroll
    for (int j = 0; j < 8; ++j) {
      out[(rbase + j) * outStride + cbase + t * 16] = acc[t][j];
    }
  }
}

// ---------------------------------------------------------------------------
// Kernel 3: windowed attention, one workgroup per (b,w), one wave per head.
// Only query row i=0 (position p = 2047-32w) is needed by the reference.
// ---------------------------------------------------------------------------
__device__ __forceinline__ float wave_max32(float v) {
#pragma unroll
  for (int off = 16; off; off >>= 1) v = fmaxf(v, __shfl_xor(v, off, 32));
  return v;
}
__device__ __forceinline__ float wave_sum32(float v) {
#pragma unroll
  for (int off = 16; off; off >>= 1) v += __shfl_xor(v, off, 32);
  return v;
}

__global__ __launch_bounds__(256) void attn_kernel(
    const float* __restrict__ kv,   // [B*2048][1024] : K cols 0..511, V 512..1023
    const float* __restrict__ qb,   // [B*64][512]
    const float* __restrict__ mask, // [B][2048]
    float* __restrict__ out,        // [B*64][512]
    float* __restrict__ mask_out) { // [B*64]
  __shared__ float qs[8][64];
  __shared__ float probs[8][64];

  const int bw = blockIdx.x;
  const int b = bw >> 6;
  const int w = bw & 63;
  const int p = 2047 - 32 * w;
  const int h = threadIdx.x >> 5;
  const int lane = threadIdx.x & 31;

  const float* qrow = qb + (b * 64 + w) * 512 + h * 64;
  qs[h][lane]      = qrow[lane];
  qs[h][lane + 32] = qrow[lane + 32];
  __syncthreads();

  const float* mrow = mask + b * 2048;
  const float mvi = mrow[p];

  // logits for keys j = lane and j = lane+32
  float logit[2];
#pragma unroll
  for (int s = 0; s < 2; ++s) {
    int j = lane + 32 * s;
    int pos = p + j;
    float dot = 0.f, mvj;
    if (pos < 2048) {
      const float* krow = kv + (b * 2048 + pos) * 1024 + h * 64;
#pragma unroll
      for (int d = 0; d < 64; d += 4) {
        float4 kd = *(const float4*)&krow[d];
        dot += kd.x * qs[h][d] + kd.y * qs[h][d + 1] +
               kd.z * qs[h][d + 2] + kd.w * qs[h][d + 3];
      }
      mvj = mrow[pos];
    } else {
      mvj = -100.f;  // padded window slot: K row is zero, mask value -100
    }
    logit[s] = dot * 0.125f - mvi * mvj;
  }

  // exact softmax across the 64 window slots
  float mx = wave_max32(fmaxf(logit[0], logit[1]));
  float e0 = expf(logit[0] - mx);
  float e1 = expf(logit[1] - mx);
  float inv = 1.f / wave_sum32(e0 + e1);
  probs[h][lane]      = e0 * inv;
  probs[h][lane + 32] = e1 * inv;
  __syncthreads();

  // out[d] = sum_j p_j * V[j][d]; padded rows are zero -> skip
  float acc0 = 0.f, acc1 = 0.f;
  const int jmax = min(64, 2048 - p);
  for (int j = 0; j < jmax; ++j) {
    float pj = probs[h][j];
    const float* vrow = kv + (b * 2048 + p + j) * 1024 + 512 + h * 64;
    acc0 += pj * vrow[lane];
    acc1 += pj * vrow[lane + 32];
  }
  float* orow = out + (b * 64 + w) * 512 + h * 64;
  orow[lane]      = acc0;
  orow[lane + 32] = acc1;

  // mask_out[b,w] = mean over 64 slots of (mask value or -100 pad)
  if (h == 0) {
    float s = ((p + lane < 2048) ? mrow[p + lane] : -100.f) +
              ((p + lane + 32 < 2048) ? mrow[p + lane + 32] : -100.f);
    float tot = wave_sum32(s);
    if (lane == 0) mask_out[b * 64 + w] = tot * (1.f / 64.f);
  }
}

// ---------------------------------------------------------------------------
extern "C" void kernel_launch(void* const* d_in, const int* in_sizes, int n_in,
                              void* d_out, int out_size, void* d_ws, size_t ws_size,
                              hipStream_t stream) {
  const float* x        = (const float*)d_in[0];  // [8,2048,512]
  const float* mask_mat = (const float*)d_in[1];  // [8,2048]
  const float* qkv_w    = (const float*)d_in[2];  // [512,1536]
  float* out = (float*)d_out;                     // 262144 + 512

  float* ws = (float*)d_ws;
  float* pe = ws;                 //  4 MB : [2048][512]
  float* kv = pe + NPOS * C_DIM;  // 64 MB : [16384][1024]  (K || V)
  float* qb = kv + (size_t)16384 * 1024;  // 1 MB : [512][512]

  // 1) PE table
  pe_kernel<<<(NPOS * C_DIM + 255) / 256, 256, 0, stream>>>(pe);

  // 2) KV = (x+PE) @ W[:,512:1536]  -> M=16384, N=1024
  gemm_pe_wmma<<<dim3(16384 / TILE_M, 1024 / TILE_N), 256, 0, stream>>>(
      x, pe, qkv_w + 512, kv, 1024, 0);

  // 3) Q at gathered window-start rows: M=512, N=512
  gemm_pe_wmma<<<dim3(512 / TILE_M, 512 / TILE_N), 256, 0, stream>>>(
      x, pe, qkv_w, qb, 512, 1);

  // 4) attention + mask_out
  attn_kernel<<<8 * 64, 256, 0, stream>>>(kv, qb, mask_mat, out,
                                          out + 8 * 64 * C_DIM);
}